// LocalResiduals_19971597927121
// MI455X (gfx1250) — compile-verified
//
#include <hip/hip_runtime.h>
#include <hip/hip_bf16.h>
#include <stdint.h>

// Problem constants (from the reference)
#define NPIX   16384
#define NFEAT  8      // NF
#define MD_    16
#define KNB    9
#define NIN    16     // NF + ND
#define KDIM   144    // KNB * NIN
#define MLPH   64
#define D0     24     // MD + NDM
#define BATCH  16     // B
#define WTILE  (KNB * MD_ * NIN)   // 2304 floats = 9216 B per pixel

#define PPW    4      // pixels per wave (one wave per block)

// LDS row strides (floats), padded for bank-conflict avoidance, even for b64 align
#define XS 148
#define MS 26
#define HS 66

typedef __attribute__((ext_vector_type(2))) float v2f;
typedef __attribute__((ext_vector_type(8))) float v8f;

__global__ __launch_bounds__(32) void LocalResiduals_kernel(
    const float* __restrict__ y,       // (16, 8, NPIX)
    const float* __restrict__ noise,   // (16, 8, NPIX)
    const float* __restrict__ noise2,  // (16, NPIX, 8)
    const float* __restrict__ wm,      // (NPIX, 9, 16, 16)
    const float* __restrict__ w1,      // (64, 24)
    const float* __restrict__ b1,      // (64)
    const float* __restrict__ w2,      // (8, 64)
    const float* __restrict__ b2,      // (8)
    const int*   __restrict__ nbr,     // (NPIX, 9) int32
    float*       __restrict__ out)     // (16, 8, NPIX)
{
    __shared__ __align__(16) float w1s[MLPH * D0];     // raw w1 (64x24)
    __shared__ __align__(16) float w2s[16 * HS];       // w2^T zero-padded [f(16)][d(64)]
    __shared__ __align__(16) float Outs[128 * PPW];    // [row(B*NF)][pixel-in-block]
    __shared__ __align__(16) float Ws[2][WTILE];       // double-buffered weight tile
    __shared__ __align__(16) float Xs[BATCH * XS];     // gathered X (16 x 144, pad)
    __shared__ __align__(16) float Ms_[BATCH * MS];    // mlp_in (16 x 24, pad)
    __shared__ __align__(16) float Hs[BATCH * HS];     // h (16 x 64, pad)
    __shared__ int Qs[12];

    const int lane = threadIdx.x;
    const int hh   = lane >> 4;          // 16-lane half
    const int mn   = lane & 15;          // A-row / B,C column index
    const int p0   = blockIdx.x * PPW;

    // ---- loop-invariant staging: w1 (b128), w2^T zero-padded, biases ----
    {
        const float4* g4 = (const float4*)w1;
        float4* s4 = (float4*)w1s;
        #pragma unroll
        for (int t = 0; t < (MLPH * D0) / 4 / 32; ++t)      // 12 iters
            s4[t * 32 + lane] = g4[t * 32 + lane];
        #pragma unroll
        for (int t = 0; t < (16 * 64) / 32; ++t) {          // 32 iters
            int idx = t * 32 + lane;
            int f = idx >> 6, d = idx & 63;
            w2s[f * HS + d] = (f < 8) ? w2[f * 64 + d] : 0.0f;
        }
    }
    float bias1[4];
    #pragma unroll
    for (int t = 0; t < 4; ++t) bias1[t] = b1[t * 16 + mn];
    float bias2 = b2[mn & 7];
    if (mn >= 8) bias2 = 0.0f;

    // ---- async weight-tile prefetch: 18 x b128 per lane-wave = 9216 B ----
    // Stream-once data -> non-temporal so it does not evict the L2-resident feats.
    auto prefetch_w = [&](int pix, int buf) {
        uint32_t lds = (uint32_t)(uintptr_t)(&Ws[buf][0]) + (uint32_t)(lane * 16);
        uint64_t ga  = (uint64_t)(uintptr_t)(wm + (size_t)(p0 + pix) * WTILE)
                     + (uint64_t)(lane * 16);
        #pragma unroll
        for (int t = 0; t < 18; ++t)
            asm volatile(
                "global_load_async_to_lds_b128 %0, %1, off offset:%2 th:TH_LOAD_NT"
                :: "v"(lds), "v"(ga), "i"(t * 512) : "memory");
    };

    prefetch_w(0, 0);

    #pragma unroll 1
    for (int i = 0; i < PPW; ++i) {
        const int p   = p0 + i;
        const int buf = i & 1;
        if (i + 1 < PPW) prefetch_w(i + 1, buf ^ 1);   // overlap next tile with compute

        // ---- gather X(16 x 144): feats of the 9 neighbors (L2-resident) ----
        if (lane < KNB) Qs[lane] = nbr[p * KNB + lane];
        for (int t = 0; t < 72; ++t) {
            int idx = t * 32 + lane;             // 0..2303
            int m   = idx / KDIM;
            int kk  = idx - m * KDIM;
            int k   = kk >> 4;
            int n   = kk & 15;
            int q   = Qs[k];
            float v = (n < NFEAT)
                ? y[((m << 3) + n) * NPIX + q]
                : noise[((m << 3) + (n - NFEAT)) * NPIX + q];
            Xs[m * XS + kk] = v;
        }

        // wait for this pixel's weight tile (allow the one just-issued prefetch)
        if (i + 1 < PPW) asm volatile("s_wait_asynccnt 18" ::: "memory");
        else             asm volatile("s_wait_asynccnt 0"  ::: "memory");

        // ============ Stage 1: D1(16x16) = X(16x144) * Wp(144x16) =========
        // A-frag (32b 16x4): lane m=mn, VGPR v, half hh -> K = kb + 2*hh + v
        // B-frag: lane n=md; K pair contiguous in nin of raw [k][md][nin]
        v8f acc1 = {};
        {
            const float* xrow = &Xs[mn * XS + 2 * hh];
            const float* wt   = &Ws[buf][mn * 16 + 2 * hh];
            #pragma unroll
            for (int kb = 0; kb < KDIM; kb += 4) {
                v2f a = *(const v2f*)(xrow + kb);
                v2f b = *(const v2f*)(wt + (kb >> 4) * 256 + (kb & 15));
                acc1 = __builtin_amdgcn_wmma_f32_16x16x4_f32(
                    false, a, false, b, (short)0, acc1, false, false);
            }
        }

        // ---- build mlp_in(16x24) = [intermediate | noise2] in LDS ----
        #pragma unroll
        for (int v = 0; v < 8; ++v)             // C layout: row M = v + 8*hh, col mn
            Ms_[(v + 8 * hh) * MS + mn] = acc1[v];
        #pragma unroll
        for (int t = 0; t < 4; ++t) {           // 128 noise2 values
            int idx = t * 32 + lane;
            int m = idx >> 3, j = idx & 7;
            Ms_[m * MS + 16 + j] = noise2[((size_t)m * NPIX + p) * 8 + j];
        }

        // ===== Stage 2: H(16x64) = relu(mlp_in(16x24) * w1^T(24x64) + b1) =
        v8f acc2[4];
        #pragma unroll
        for (int t = 0; t < 4; ++t) {
            float bv = bias1[t];
            acc2[t] = (v8f){bv, bv, bv, bv, bv, bv, bv, bv};
        }
        #pragma unroll
        for (int kb = 0; kb < D0; kb += 4) {
            v2f a = *(const v2f*)(&Ms_[mn * MS + kb + 2 * hh]);
            #pragma unroll
            for (int t = 0; t < 4; ++t) {
                v2f b = *(const v2f*)(&w1s[(t * 16 + mn) * D0 + kb + 2 * hh]);
                acc2[t] = __builtin_amdgcn_wmma_f32_16x16x4_f32(
                    false, a, false, b, (short)0, acc2[t], false, false);
            }
        }
        #pragma unroll
        for (int t = 0; t < 4; ++t)
            #pragma unroll
            for (int v = 0; v < 8; ++v)
                Hs[(v + 8 * hh) * HS + t * 16 + mn] = fmaxf(acc2[t][v], 0.0f);

        // ===== Stage 3: OUT(16x8) = H(16x64) * w2^T(64x8pad16) + b2 =======
        v8f acc3 = (v8f){bias2, bias2, bias2, bias2, bias2, bias2, bias2, bias2};
        #pragma unroll
        for (int kb = 0; kb < MLPH; kb += 4) {
            v2f a = *(const v2f*)(&Hs[mn * HS + kb + 2 * hh]);
            v2f b = *(const v2f*)(&w2s[mn * HS + kb + 2 * hh]);
            acc3 = __builtin_amdgcn_wmma_f32_16x16x4_f32(
                false, a, false, b, (short)0, acc3, false, false);
        }

        // ---- stash this pixel's column for coalesced stores ----
        if (mn < 8) {
            #pragma unroll
            for (int v = 0; v < 8; ++v) {
                int m = v + 8 * hh;
                Outs[(m * 8 + mn) * PPW + i] = acc3[v];
            }
        }
    }

    // ---- coalesced 16B stores: 128 rows x 4 consecutive pixels ----
    #pragma unroll
    for (int t = 0; t < 4; ++t) {
        int rr = t * 32 + lane;                 // row in (B*NF)
        float4 val = *(const float4*)(&Outs[rr * PPW]);
        *(float4*)(&out[(size_t)rr * NPIX + p0]) = val;
    }
}

extern "C" void kernel_launch(void* const* d_in, const int* in_sizes, int n_in,
                              void* d_out, int out_size, void* d_ws, size_t ws_size,
                              hipStream_t stream) {
    (void)in_sizes; (void)n_in; (void)out_size; (void)d_ws; (void)ws_size;
    const float* y      = (const float*)d_in[0];
    const float* noise  = (const float*)d_in[1];
    const float* noise2 = (const float*)d_in[2];
    const float* wm     = (const float*)d_in[3];
    const float* w1     = (const float*)d_in[4];
    const float* b1     = (const float*)d_in[5];
    const float* w2     = (const float*)d_in[6];
    const float* b2     = (const float*)d_in[7];
    const int*   nbr    = (const int*)d_in[8];
    float* outp = (float*)d_out;

    dim3 grid(NPIX / PPW), block(32);
    LocalResiduals_kernel<<<grid, block, 0, stream>>>(
        y, noise, noise2, wm, w1, b1, w2, b2, nbr, outp);
}